// ProbabilityMatrixFactorization_7576322310165
// MI455X (gfx1250) — compile-verified
//
#include <hip/hip_runtime.h>
#include <hip/hip_bf16.h>

// CDNA5 / gfx1250, wave32.
typedef float v2f __attribute__((ext_vector_type(2)));
typedef float v8f __attribute__((ext_vector_type(8)));

#define TILE      128
#define KDIM      64
#define LDS_PAD   68   // row stride in floats: (68*r + c) % 64 = (4r+c)%64 -> conflict-free,
                       // and 68*r + c4 is a multiple of 4 floats -> 16B-aligned b128 stores.

__global__ __launch_bounds__(256)
void pmf_gemm_kernel(const int*   __restrict__ user_ids,
                     const int*   __restrict__ item_ids,
                     const float* __restrict__ user_weight,
                     const float* __restrict__ item_weight,
                     const float* __restrict__ user_bias,
                     const float* __restrict__ item_bias,
                     const float* __restrict__ bias,
                     float*       __restrict__ out,
                     int n_items_batch)
{
    __shared__ float A_s[TILE * LDS_PAD];   // gathered user rows, [row][k]
    __shared__ float B_s[TILE * LDS_PAD];   // gathered item rows, [row][k]
    __shared__ float ub_s[TILE];
    __shared__ float ib_s[TILE];

    const int m0   = blockIdx.y * TILE;
    const int n0   = blockIdx.x * TILE;
    const int tid  = threadIdx.x;        // 0..255
    const int lane = tid & 31;
    const int wave = tid >> 5;           // 0..7

    // ---------------- Gather phase: global -> LDS ----------------
    // 128 rows x 16 float4 per matrix = 2048 float4; 256 threads x 8 iters.
#pragma unroll
    for (int i = 0; i < 8; ++i) {
        const int j  = tid + i * 256;        // 0..2047
        const int r  = j >> 4;               // row 0..127
        const int c4 = (j & 15) << 2;        // col 0,4,...,60
        const int urow = user_ids[m0 + r];
        const float4 av = *(const float4*)(user_weight + (size_t)urow * KDIM + c4);
        *(float4*)(&A_s[r * LDS_PAD + c4]) = av;
        const int irow = item_ids[n0 + r];
        const float4 bv = *(const float4*)(item_weight + (size_t)irow * KDIM + c4);
        *(float4*)(&B_s[r * LDS_PAD + c4]) = bv;
    }
    if (tid < TILE) {
        ub_s[tid] = user_bias[user_ids[m0 + tid]];
    } else {
        const int t = tid - TILE;
        ib_s[t] = item_bias[item_ids[n0 + t]];
    }
    __syncthreads();

    // ---------------- Compute phase: WMMA f32 16x16x4 ----------------
    // Wave grid: 4 waves along M (32 rows each), 2 waves along N (64 cols each).
    const int wm = wave & 3;                 // M wave: rows  wm*32 .. wm*32+31
    const int wn = wave >> 2;                // N wave: cols  wn*64 .. wn*64+63

    v8f acc[2][4];
#pragma unroll
    for (int mi = 0; mi < 2; ++mi)
#pragma unroll
        for (int ni = 0; ni < 4; ++ni)
            acc[mi][ni] = v8f{0.f, 0.f, 0.f, 0.f, 0.f, 0.f, 0.f, 0.f};

    // ISA 16x4 f32 A layout: lanes 0-15 = M rows, VGPR0/1 hold K={0,1} (lanes 0-15)
    // and K={2,3} (lanes 16-31).  B (4x16) mirrors it: lane = N, VGPR pair = K.
    const int l15  = lane & 15;
    const int koff = (lane >> 4) << 1;       // 0 or 2

    const float* Abase = &A_s[(wm * 32 + l15) * LDS_PAD];
    const float* Bbase = &B_s[(wn * 64 + l15) * LDS_PAD];

#pragma unroll
    for (int k = 0; k < KDIM; k += 4) {
        const v2f a0 = *(const v2f*)(Abase + k + koff);
        const v2f a1 = *(const v2f*)(Abase + 16 * LDS_PAD + k + koff);
        const v2f b0 = *(const v2f*)(Bbase + k + koff);
        const v2f b1 = *(const v2f*)(Bbase + 16 * LDS_PAD + k + koff);
        const v2f b2 = *(const v2f*)(Bbase + 32 * LDS_PAD + k + koff);
        const v2f b3 = *(const v2f*)(Bbase + 48 * LDS_PAD + k + koff);

        acc[0][0] = __builtin_amdgcn_wmma_f32_16x16x4_f32(false, a0, false, b0, (short)0, acc[0][0], false, false);
        acc[0][1] = __builtin_amdgcn_wmma_f32_16x16x4_f32(false, a0, false, b1, (short)0, acc[0][1], false, false);
        acc[0][2] = __builtin_amdgcn_wmma_f32_16x16x4_f32(false, a0, false, b2, (short)0, acc[0][2], false, false);
        acc[0][3] = __builtin_amdgcn_wmma_f32_16x16x4_f32(false, a0, false, b3, (short)0, acc[0][3], false, false);
        acc[1][0] = __builtin_amdgcn_wmma_f32_16x16x4_f32(false, a1, false, b0, (short)0, acc[1][0], false, false);
        acc[1][1] = __builtin_amdgcn_wmma_f32_16x16x4_f32(false, a1, false, b1, (short)0, acc[1][1], false, false);
        acc[1][2] = __builtin_amdgcn_wmma_f32_16x16x4_f32(false, a1, false, b2, (short)0, acc[1][2], false, false);
        acc[1][3] = __builtin_amdgcn_wmma_f32_16x16x4_f32(false, a1, false, b3, (short)0, acc[1][3], false, false);
    }

    // ---------------- Epilogue: + ub + ib + bias, store ----------------
    // C/D 16x16 f32 layout: VGPR j holds M=j (lanes 0-15) and M=j+8 (lanes 16-31);
    // N = lane & 15.
    const float gb   = bias[0];
    const int rhalf  = (lane >> 4) << 3;     // 0 or 8
    const size_t ldo = (size_t)n_items_batch;

#pragma unroll
    for (int mi = 0; mi < 2; ++mi) {
#pragma unroll
        for (int ni = 0; ni < 4; ++ni) {
            const int row_t = wm * 32 + mi * 16 + rhalf;   // tile-local row of VGPR 0
            const int col_t = wn * 64 + ni * 16 + l15;     // tile-local col
            const float cb  = ib_s[col_t] + gb;
            float* optr = out + (size_t)(m0 + row_t) * ldo + (n0 + col_t);
#pragma unroll
            for (int j = 0; j < 8; ++j) {
                optr[(size_t)j * ldo] = acc[mi][ni][j] + ub_s[row_t + j] + cb;
            }
        }
    }
}

extern "C" void kernel_launch(void* const* d_in, const int* in_sizes, int n_in,
                              void* d_out, int out_size, void* d_ws, size_t ws_size,
                              hipStream_t stream) {
    const int*   user_ids    = (const int*)  d_in[0];
    const int*   item_ids    = (const int*)  d_in[1];
    const float* user_weight = (const float*)d_in[2];
    const float* item_weight = (const float*)d_in[3];
    const float* user_bias   = (const float*)d_in[4];
    const float* item_bias   = (const float*)d_in[5];
    const float* bias        = (const float*)d_in[6];
    float*       out         = (float*)      d_out;

    const int Bu = in_sizes[0];   // 4096
    const int Bi = in_sizes[1];   // 4096

    dim3 grid(Bi / TILE, Bu / TILE);   // 32 x 32 tiles of 128x128
    pmf_gemm_kernel<<<grid, 256, 0, stream>>>(user_ids, item_ids,
                                              user_weight, item_weight,
                                              user_bias, item_bias, bias,
                                              out, Bi);
}